// Net_65609920413817
// MI455X (gfx1250) — compile-verified
//
#include <hip/hip_runtime.h>
#include <math.h>

// ---------------------------------------------------------------------------
// Canny edge pipeline for MI455X (gfx1250, wave32).
// Memory-bound stencil: fused gauss->sobel->mag->NMS in one LDS-tiled pass,
// global-max reduction fused via atomics, then a vectorized threshold pass.
// Tile loads use CDNA5 async global->LDS (ASYNCcnt): b128 wide path for
// interior tiles, clamped b32 path for image-edge tiles.
// ---------------------------------------------------------------------------

#define TILE 32
#define HALO 3
#define XT   (TILE + 2 * HALO)  // 38 rows of x tile (halo 3)
#define XW   40                 // 40 cols of x tile, origin ox-4 (16B aligned)
#define GT   (TILE + 4)         // 36: gaussian tile with halo 2
#define ST   (TILE + 2)         // 34: sobel/mag tile with halo 1
#define PITCH 40                // LDS pitch (non-pow2 -> bank friendly)

#if defined(__AMDGCN__) && defined(__gfx1250__) && \
    __has_builtin(__builtin_amdgcn_global_load_async_to_lds_b32) && \
    __has_builtin(__builtin_amdgcn_s_wait_asynccnt)
#define USE_ASYNC_LDS 1
// addrspace-qualified pointee types matching the builtin signatures:
// b32 takes int* (AS1/AS3); b128 takes 4-int-vector* (AS1/AS3).
typedef __attribute__((address_space(1))) int async_gptr_t;   // global, b32
typedef __attribute__((address_space(3))) int async_lptr_t;   // LDS, b32
typedef int v4i_native __attribute__((vector_size(16)));
typedef __attribute__((address_space(1))) v4i_native async_gptr4_t;  // global, b128
typedef __attribute__((address_space(3))) v4i_native async_lptr4_t;  // LDS, b128
#if __has_builtin(__builtin_amdgcn_global_load_async_to_lds_b128)
#define HAVE_ASYNC_B128 1
#else
#define HAVE_ASYNC_B128 0
#endif
#else
#define USE_ASYNC_LDS 0
#define HAVE_ASYNC_B128 0
#endif

typedef float v4f __attribute__((ext_vector_type(4)));

__device__ __forceinline__ int iclampi(int v, int hi) {
    v = v < 0 ? 0 : v;
    return v > hi ? hi : v;
}

__device__ __forceinline__ float stable_sigmoid(float z) {
    if (z >= 0.0f) {
        float e = __expf(-z);
        return 1.0f / (1.0f + e);
    }
    float e = __expf(z);
    return e / (1.0f + e);
}

// ---------------------------------------------------------------------------
// Pass 1: fused gauss + sobel + magnitude + non-max suppression per 32x32 tile
// Writes: mag, thin_raw (unnormalized); atomicMax of thin_raw into *gmax.
// Replicate padding == evaluating each conv stage at coordinate-clamped
// positions, applied independently per stage (matches two-stage 'edge' pad).
// sX column c maps to global column (ox - 4 + c), clamped on edge tiles.
// ---------------------------------------------------------------------------
__global__ __launch_bounds__(256) void canny_stencil(
    const float* __restrict__ x,
    const float* __restrict__ wg,
    const float* __restrict__ wsx,
    const float* __restrict__ wsy,
    float* __restrict__ out_mag,
    float* __restrict__ out_thin,
    unsigned int* __restrict__ gmax,
    int H, int W)
{
    __shared__ float sX[XT][XW];
    __shared__ float sG[GT][PITCH];
    __shared__ float sSx[ST][PITCH];
    __shared__ float sSy[ST][PITCH];
    __shared__ float sM[ST][PITCH];
    __shared__ float sRed[8];

    const int tx  = threadIdx.x;            // 0..31 (one wave32 per ty)
    const int ty  = threadIdx.y;            // 0..7
    const int lid = ty * 32 + tx;
    const int ox  = blockIdx.x * TILE;
    const int oy  = blockIdx.y * TILE;
    const size_t base = (size_t)blockIdx.z * (size_t)H * (size_t)W;

    // Uniform 3x3 kernels -> scalar regs (SMEM loads, overlap with async DMA).
    float g[9], kx[9], ky[9];
#pragma unroll
    for (int i = 0; i < 9; ++i) {
        g[i]  = wg[i];
        kx[i] = wsx[i];
        ky[i] = wsy[i];
    }

    // ---- Stage 0: x tile (rows halo 3, cols [ox-4, ox+36)) -> LDS ----
    const bool interiorTile =
        (ox >= 4) && (ox + 36 <= W) && (oy >= HALO) && (oy + XT - HALO <= H);

#if USE_ASYNC_LDS && HAVE_ASYNC_B128
    if (interiorTile) {
        // Wide path: 38 rows x 10 x b128 (16B-aligned: (ox-4)*4 % 16 == 0).
        for (int i = lid; i < XT * 10; i += 256) {
            int r = i / 10;
            int q = i - r * 10;
            const float* gp = x + base + (size_t)(oy - HALO + r) * W + (ox - 4) + q * 4;
            __builtin_amdgcn_global_load_async_to_lds_b128(
                (async_gptr4_t*)gp, (async_lptr4_t*)&sX[r][q * 4], 0, 0);
        }
    } else
#endif
    {
        for (int i = lid; i < XT * XW; i += 256) {
            int r  = i / XW;
            int c  = i - r * XW;
            int gy = iclampi(oy - HALO + r, H - 1);
            int gxc = iclampi(ox - 4 + c, W - 1);
            const float* gp = x + base + (size_t)gy * W + gxc;
#if USE_ASYNC_LDS
            __builtin_amdgcn_global_load_async_to_lds_b32(
                (async_gptr_t*)gp, (async_lptr_t*)&sX[r][c], 0, 0);
#else
            sX[r][c] = *gp;
#endif
        }
    }
#if USE_ASYNC_LDS
    __builtin_amdgcn_s_wait_asynccnt(0);
#endif
    __syncthreads();

    // ---- Stage 1: gaussian blur, tile GTxGT (halo 2) ----
    for (int i = lid; i < GT * GT; i += 256) {
        int r  = i / GT;
        int c  = i - r * GT;
        int cy = iclampi(oy - 2 + r, H - 1);
        int cx = iclampi(ox - 2 + c, W - 1);
        int ry[3], rx[3];
#pragma unroll
        for (int a = 0; a < 3; ++a) {
            ry[a] = iclampi(cy - 1 + a, H - 1) - (oy - HALO);
            rx[a] = iclampi(cx - 1 + a, W - 1) - (ox - 4);
        }
        float acc = 0.0f;
#pragma unroll
        for (int a = 0; a < 3; ++a)
#pragma unroll
            for (int b = 0; b < 3; ++b)
                acc = fmaf(g[a * 3 + b], sX[ry[a]][rx[b]], acc);
        sG[r][c] = acc;
    }
    __syncthreads();

    // ---- Stage 2: sobel x/y + magnitude, tile STxST (halo 1) ----
    const float eps = 1e-10f;
    for (int i = lid; i < ST * ST; i += 256) {
        int r  = i / ST;
        int c  = i - r * ST;
        int cy = iclampi(oy - 1 + r, H - 1);
        int cx = iclampi(ox - 1 + c, W - 1);
        int ry[3], rx[3];
#pragma unroll
        for (int a = 0; a < 3; ++a) {
            ry[a] = iclampi(cy - 1 + a, H - 1) - (oy - 2);
            rx[a] = iclampi(cx - 1 + a, W - 1) - (ox - 2);
        }
        float ax = 0.0f, ay = 0.0f;
#pragma unroll
        for (int a = 0; a < 3; ++a)
#pragma unroll
            for (int b = 0; b < 3; ++b) {
                float v = sG[ry[a]][rx[b]];
                ax = fmaf(kx[a * 3 + b], v, ax);
                ay = fmaf(ky[a * 3 + b], v, ay);
            }
        sSx[r][c] = ax;
        sSy[r][c] = ay;
        float mx = ax + eps;
        float my = ay + eps;
        sM[r][c] = sqrtf(fmaf(mx, mx, my * my));
    }
    __syncthreads();

    // ---- Stage 3: NMS, write mag/thin, block max ----
    float vmax = 0.0f;
#pragma unroll
    for (int k = 0; k < 4; ++k) {
        int r  = ty + k * 8;   // 0..31
        int c  = tx;           // 0..31
        int gy = oy + r;
        int gx = ox + c;

        float m  = sM[r + 1][c + 1];
        float vx = sSx[r + 1][c + 1];
        float vy = sSy[r + 1][c + 1];
        float ai = fabsf(vy);   // abs of sobel_y (reference's abs_isobel)
        float aj = fabsf(vx);   // abs of sobel_x

        bool interior = (gy >= 1) && (gy <= H - 2) && (gx >= 1) && (gx <= W - 2);
        bool eroded = interior && (m > 0.0f);
        bool is_h = eroded && (ai >= aj);
        bool is_v = eroded && (ai <= aj);
        bool up    = vx >= 0.0f;
        bool down  = vx <= 0.0f;
        bool right = vy >= 0.0f;
        bool left  = vy <= 0.0f;
        bool pa = (up && right) || (down && left);
        bool pb = (down && right) || (up && left);
        float wh = aj / ai;  // raw div: inf/nan masked out by NaN-compare==false
        float wv = ai / aj;

        bool o = false;
#define MM(dr, dc) sM[r + 1 + (dr)][c + 1 + (dc)]
#define NMS_CASE(w, r1, c1, r2, c2)                                            \
        ((fmaf(MM(r2, c2), (w), MM(r1, c1) * (1.0f - (w))) <= m) &&            \
         (fmaf(MM(-(r2), -(c2)), (w), MM(-(r1), -(c1)) * (1.0f - (w))) <= m))
        // ordered overwrite semantics, same order as reference
        if (pa && is_h) o = NMS_CASE(wh,  1, 0,  1, 1);
        if (pa && is_v) o = NMS_CASE(wv,  0, 1,  1, 1);
        if (pb && is_v) o = NMS_CASE(wv,  0, 1, -1, 1);
        if (pb && is_h) o = NMS_CASE(wh, -1, 0, -1, 1);
#undef NMS_CASE
#undef MM

        size_t idx = base + (size_t)gy * W + gx;
        __builtin_nontemporal_store(m, &out_mag[idx]);  // streaming output
        float thin = o ? m : 0.0f;
        out_thin[idx] = thin;                           // re-read by pass 2: keep RT
        vmax = fmaxf(vmax, thin);
    }

    // wave32 reduce then cross-wave reduce then one atomic per block
#pragma unroll
    for (int off = 16; off > 0; off >>= 1)
        vmax = fmaxf(vmax, __shfl_xor(vmax, off, 32));
    if (tx == 0) sRed[ty] = vmax;
    __syncthreads();
    if (lid == 0) {
        float bm = sRed[0];
#pragma unroll
        for (int i = 1; i < 8; ++i) bm = fmaxf(bm, sRed[i]);
        atomicMax(gmax, __float_as_uint(bm));  // all values >= 0: bit order == float order
    }
}

// ---------------------------------------------------------------------------
// Pass 2: normalize thin by global max, compute hysteresis sigmoid response.
// float4-vectorized; non-temporal final stores.
// ---------------------------------------------------------------------------
__global__ __launch_bounds__(256) void canny_threshold(
    float* __restrict__ thin,
    float* __restrict__ res,
    const unsigned int* __restrict__ gmax,
    const float* __restrict__ plo,
    const float* __restrict__ phi,
    long long n4)
{
    long long i = (long long)blockIdx.x * 256 + threadIdx.x;
    if (i >= n4) return;

    const float maxv  = __uint_as_float(*gmax);
    const float lo    = plo[0];
    const float hi    = phi[0];
    const float alpha = 100000.0f;

    v4f t = ((const v4f*)thin)[i];
    v4f tn, r;
#pragma unroll
    for (int c = 0; c < 4; ++c) {
        float tv = t[c] / maxv;
        float strong = stable_sigmoid(alpha * (tv - hi));
        float weak   = 0.5f * stable_sigmoid(alpha * (hi - tv))
                     - 0.5f * stable_sigmoid(alpha * (lo - tv));
        tn[c] = tv;
        r[c]  = weak + strong;
    }
    __builtin_nontemporal_store(tn, ((v4f*)thin) + i);
    __builtin_nontemporal_store(r,  ((v4f*)res) + i);
}

// ---------------------------------------------------------------------------
extern "C" void kernel_launch(void* const* d_in, const int* in_sizes, int n_in,
                              void* d_out, int out_size, void* d_ws, size_t ws_size,
                              hipStream_t stream) {
    const float* x   = (const float*)d_in[0];
    const float* wg  = (const float*)d_in[1];
    const float* wsx = (const float*)d_in[2];
    const float* wst = (const float*)d_in[3];
    const float* lo  = (const float*)d_in[4];
    const float* hi  = (const float*)d_in[5];

    const int H = 2048, W = 2048;
    const int B = in_sizes[0] / (H * W);          // = 4
    const size_t N = (size_t)B * H * W;

    float* out_mag  = (float*)d_out;
    float* out_thin = out_mag + N;
    float* out_res  = out_thin + N;
    unsigned int* pmax = (unsigned int*)d_ws;

    (void)hipMemsetAsync(d_ws, 0, sizeof(unsigned int), stream);

    dim3 grid(W / TILE, H / TILE, B);
    dim3 block(32, 8);
    canny_stencil<<<grid, block, 0, stream>>>(x, wg, wsx, wst,
                                              out_mag, out_thin, pmax, H, W);

    long long n4 = (long long)(N / 4);
    long long nblk = (n4 + 255) / 256;
    canny_threshold<<<dim3((unsigned)nblk), dim3(256), 0, stream>>>(
        out_thin, out_res, pmax, lo, hi, n4);
}